// CG_8624294330573
// MI455X (gfx1250) — compile-verified
//
#include <hip/hip_runtime.h>

typedef __attribute__((ext_vector_type(2))) float v2f;
typedef __attribute__((ext_vector_type(4))) float v4f;
typedef __attribute__((ext_vector_type(8))) float v8f;

#define HFEAT 128
#define BFEAT 64
#define KTOT  320           // 2H + B
#define LDS_STRIDE 324      // dwords; 324 % 64 == 4 -> conflict-free b64 LDS reads
#define NEG_SLOPE 0.01f

__global__ void CG_zero_out(float* __restrict__ out, int n4) {
    int i = blockIdx.x * blockDim.x + threadIdx.x;
    if (i < n4) ((v4f*)out)[i] = (v4f){0.f, 0.f, 0.f, 0.f};
}

__global__ __launch_bounds__(256) void CG_edge_gemm_scatter(
    const float* __restrict__ feat,   // [N, 128]
    const float* __restrict__ gdf,    // [E, 64]
    const float* __restrict__ W,      // [128, 320] row-major (n, k)
    const float* __restrict__ bias,   // [128]
    const int*   __restrict__ src,    // [E]
    const int*   __restrict__ dst,    // [E]
    float*       __restrict__ out,    // [N, 128]
    int n_edges)
{
    extern __shared__ float sW[];     // [128][LDS_STRIDE] padded copy of W

    // Stage W into LDS (coalesced read, padded-stride write)
    for (int idx = threadIdx.x; idx < HFEAT * KTOT; idx += 256) {
        int n = idx / KTOT;
        int k = idx - n * KTOT;
        sW[n * LDS_STRIDE + k] = W[idx];
    }
    __syncthreads();

    const int lane = threadIdx.x & 31;
    const int wave = threadIdx.x >> 5;
    const int m    = lane & 15;       // edge row within tile (lanes 0-15 / 16-31 mirror)
    const int hi   = lane >> 4;       // K-half select for A/B fragments
    const int ebase = (blockIdx.x * 8 + wave) * 16;

    // Per-lane gather row for A fragments (clamped for tail safety; writes masked later)
    int eA = ebase + m;
    int eC = eA < n_edges ? eA : (n_edges - 1);
    const float* pa = feat + (size_t)src[eC] * HFEAT + 2 * hi;   // cols   0..127
    const float* pb = feat + (size_t)dst[eC] * HFEAT + 2 * hi;   // cols 128..255
    const float* pg = gdf  + (size_t)eC * BFEAT     + 2 * hi;    // cols 256..319

    v8f acc[8];
    #pragma unroll
    for (int nt = 0; nt < 8; ++nt) acc[nt] = (v8f){0.f,0.f,0.f,0.f,0.f,0.f,0.f,0.f};

    const float* sWl = sW + (size_t)m * LDS_STRIDE + 2 * hi;     // lane's B base (n = nt*16+m)

    // ---- K region 1: src features (k = 0..127) ----
    for (int ks = 0; ks < 32; ++ks) {
        v2f a = *(const v2f*)(pa + 4 * ks);
        #pragma unroll
        for (int nt = 0; nt < 8; ++nt) {
            v2f b = *(const v2f*)(sWl + (size_t)nt * 16 * LDS_STRIDE + 4 * ks);
            acc[nt] = __builtin_amdgcn_wmma_f32_16x16x4_f32(
                false, a, false, b, (short)0, acc[nt], false, false);
        }
    }
    // ---- K region 2: dst features (k = 128..255) ----
    for (int ks = 0; ks < 32; ++ks) {
        v2f a = *(const v2f*)(pb + 4 * ks);
        #pragma unroll
        for (int nt = 0; nt < 8; ++nt) {
            v2f b = *(const v2f*)(sWl + (size_t)nt * 16 * LDS_STRIDE + 128 + 4 * ks);
            acc[nt] = __builtin_amdgcn_wmma_f32_16x16x4_f32(
                false, a, false, b, (short)0, acc[nt], false, false);
        }
    }
    // ---- K region 3: edge features (k = 256..319) ----
    for (int ks = 0; ks < 16; ++ks) {
        v2f a = *(const v2f*)(pg + 4 * ks);
        #pragma unroll
        for (int nt = 0; nt < 8; ++nt) {
            v2f b = *(const v2f*)(sWl + (size_t)nt * 16 * LDS_STRIDE + 256 + 4 * ks);
            acc[nt] = __builtin_amdgcn_wmma_f32_16x16x4_f32(
                false, a, false, b, (short)0, acc[nt], false, false);
        }
    }

    // D layout (16x16 f32): lanes 0-15 -> M=v, N=lane ; lanes 16-31 -> M=8+v, N=lane-16
    int dsts[8];
    bool live[8];
    #pragma unroll
    for (int v = 0; v < 8; ++v) {
        int ev = ebase + 8 * hi + v;
        live[v] = ev < n_edges;
        dsts[v] = dst[live[v] ? ev : (n_edges - 1)];
    }

    #pragma unroll
    for (int nt = 0; nt < 8; ++nt) {
        float bv = bias[nt * 16 + m];
        #pragma unroll
        for (int v = 0; v < 8; ++v) {
            float z  = acc[nt][v] + bv;
            float lr = z > 0.f ? z : NEG_SLOPE * z;
            float sg = 1.f / (1.f + __expf(-z));
            float msg = sg * lr;
            if (live[v])
                unsafeAtomicAdd(out + (size_t)dsts[v] * HFEAT + nt * 16 + m, msg);
        }
    }
}

extern "C" void kernel_launch(void* const* d_in, const int* in_sizes, int n_in,
                              void* d_out, int out_size, void* d_ws, size_t ws_size,
                              hipStream_t stream) {
    const float* feat = (const float*)d_in[0];
    const float* gdf  = (const float*)d_in[1];
    const float* W    = (const float*)d_in[2];
    const float* bias = (const float*)d_in[3];
    const int*   src  = (const int*)d_in[4];
    const int*   dst  = (const int*)d_in[5];
    float* out = (float*)d_out;

    int n_edges = in_sizes[4];

    // Zero-init output (poisoned by harness; we accumulate atomically)
    int n4 = out_size / 4;
    CG_zero_out<<<(n4 + 255) / 256, 256, 0, stream>>>(out, n4);

    // 8 waves/block, 16 edges/wave -> 128 edges per block
    int blocks = (n_edges + 127) / 128;
    size_t lds_bytes = (size_t)HFEAT * LDS_STRIDE * sizeof(float);  // 162 KB < 320 KB/WGP
    CG_edge_gemm_scatter<<<blocks, 256, lds_bytes, stream>>>(
        feat, gdf, W, bias, src, dst, out, n_edges);
}